// GraphLevelVAE_5265629905237
// MI455X (gfx1250) — compile-verified
//
#include <hip/hip_runtime.h>
#include <hip/hip_bf16.h>

// ---------------- problem constants ----------------
#define Bsz  16
#define Nn   256
#define INF_ 64
#define HID  128
#define LAT  32
#define NT   4096      // Bsz*Nn
#define NE   65536     // NT*DEG
#define BNN  1048576   // Bsz*Nn*Nn

typedef __attribute__((ext_vector_type(2))) float v2f;
typedef __attribute__((ext_vector_type(8))) float v8f;

// =====================================================================
// f32 WMMA GEMM:  C[M,Ncols] = A[M,K] @ W[:,koff:koff+K]^T
//   A row-major, lda;  W row-major [Ncols rows, ldw], column offset koff
//   One wave (32 threads) computes one 16x16 tile via V_WMMA_F32_16X16X4_F32.
//   grid = (Ncols/16, M/16), block = 32. EXEC is all ones (no divergence).
// =====================================================================
__global__ void gemm_wmma_f32(const float* __restrict__ A, int lda,
                              const float* __restrict__ W, int ldw, int koff,
                              float* __restrict__ C, int ldc, int K) {
    const int lane = threadIdx.x;            // 0..31
    const int l15  = lane & 15;
    const int hi   = lane >> 4;              // 0 or 1
    const int row  = blockIdx.y * 16 + l15;  // A row (M index)
    const int col  = blockIdx.x * 16 + l15;  // W row (N index)
    const int kh   = hi * 2;                 // lanes 16..31 hold K=2,3

    v8f acc = {};
    for (int kb = 0; kb < K; kb += 4) {
        v2f a, b;
        a.x = A[row * lda + kb + kh];
        a.y = A[row * lda + kb + kh + 1];
        b.x = W[col * ldw + koff + kb + kh];
        b.y = W[col * ldw + koff + kb + kh + 1];
        acc = __builtin_amdgcn_wmma_f32_16x16x4_f32(
            /*neg_a=*/false, a, /*neg_b=*/false, b,
            /*c_mod=*/(short)0, acc, /*reuse_a=*/false, /*reuse_b=*/false);
    }
    const int r0 = blockIdx.y * 16 + hi * 8; // D: VGPR r -> rows r / r+8
#pragma unroll
    for (int r = 0; r < 8; ++r)
        C[(r0 + r) * ldc + col] = acc[r];
}

// ---------------- degree / normalization ----------------
__global__ void k_deg_init(float* deg) {
    int i = blockIdx.x * blockDim.x + threadIdx.x;
    if (i < NT) deg[i] = 1.0f;               // self loop
}
__global__ void k_deg_acc(const int* __restrict__ dst, float* deg) {
    int e = blockIdx.x * blockDim.x + threadIdx.x;
    if (e < NE) atomicAdd(&deg[dst[e]], 1.0f);
}
__global__ void k_dinv(float* deg) {
    int i = blockIdx.x * blockDim.x + threadIdx.x;
    if (i < NT) deg[i] = rsqrtf(deg[i]);
}

// ---------------- GCN aggregation ----------------
__global__ void k_agg_init(const float* __restrict__ h, const float* __restrict__ dinv,
                           const float* __restrict__ bias, float* __restrict__ agg) {
    int i = blockIdx.x * blockDim.x + threadIdx.x;
    if (i >= NT * HID) return;
    int n = i >> 7, c = i & 127;
    float dv = dinv[n];
    agg[i] = bias[c] + dv * dv * h[i];       // bias + self-loop term
}
__global__ void k_agg_edges(const float* __restrict__ h, const float* __restrict__ dinv,
                            const int* __restrict__ src, const int* __restrict__ dst,
                            float* __restrict__ agg) {
    int e = blockIdx.x;                      // one block per edge
    int c = threadIdx.x;                     // 128 channels
    int s = src[e], d = dst[e];
    float w = dinv[s] * dinv[d];
    atomicAdd(&agg[d * HID + c], w * h[s * HID + c]);
}

// ---------------- BatchNorm (training stats) + leakyReLU(0.2) ----------------
__global__ void k_bn_lrelu(const float* __restrict__ agg, const float* __restrict__ gamma,
                           const float* __restrict__ beta, float* __restrict__ out) {
    __shared__ float s1[256], s2[256];
    __shared__ float s_scale, s_shift;
    int c = blockIdx.x, t = threadIdx.x;
    float a = 0.0f, b = 0.0f;
    for (int n = t; n < NT; n += 256) {
        float v = agg[n * HID + c];
        a += v; b += v * v;
    }
    s1[t] = a; s2[t] = b; __syncthreads();
    for (int off = 128; off > 0; off >>= 1) {
        if (t < off) { s1[t] += s1[t + off]; s2[t] += s2[t + off]; }
        __syncthreads();
    }
    if (t == 0) {
        float mu  = s1[0] * (1.0f / NT);
        float var = s2[0] * (1.0f / NT) - mu * mu;   // biased variance
        float inv = rsqrtf(var + 1e-5f) * gamma[c];
        s_scale = inv;
        s_shift = beta[c] - mu * inv;
    }
    __syncthreads();
    float sc = s_scale, sh = s_shift;
    for (int n = t; n < NT; n += 256) {
        float v = agg[n * HID + c] * sc + sh;
        out[n * HID + c] = (v > 0.0f) ? v : 0.2f * v;
    }
}

// ---------------- global mean pool ----------------
__global__ void k_pool(const float* __restrict__ h, float* __restrict__ g) {
    int i = blockIdx.x * blockDim.x + threadIdx.x;
    if (i >= Bsz * HID) return;
    int b = i >> 7, c = i & 127;
    float s = 0.0f;
    for (int n = 0; n < Nn; ++n) s += h[(b * Nn + n) * HID + c];
    g[i] = s * (1.0f / Nn);
}

// ---------------- mu / logvar / z (reparameterization) ----------------
__global__ void k_muz(const float* __restrict__ g,
                      const float* __restrict__ Wmu, const float* __restrict__ bmu,
                      const float* __restrict__ Wlv, const float* __restrict__ blv,
                      const float* __restrict__ eps,
                      float* __restrict__ mu_out, float* __restrict__ lv_out,
                      float* __restrict__ z) {
    int i = blockIdx.x * blockDim.x + threadIdx.x;
    if (i >= Bsz * LAT) return;
    int b = i >> 5, l = i & 31;
    float m = bmu[l], v = blv[l];
    for (int k = 0; k < HID; ++k) {
        float gv = g[b * HID + k];
        m = fmaf(gv, Wmu[l * HID + k], m);
        v = fmaf(gv, Wlv[l * HID + k], v);
    }
    mu_out[i] = m;
    lv_out[i] = v;
    z[i] = m + expf(0.5f * v) * eps[i];
}

// ---------------- decoder MLP on z ----------------
__global__ void k_hg1(const float* __restrict__ z, const float* __restrict__ Wd1,
                      const float* __restrict__ bd1, float* __restrict__ hg1) {
    int i = blockIdx.x * blockDim.x + threadIdx.x;
    if (i >= Bsz * HID) return;
    int b = i >> 7, h = i & 127;
    float t = bd1[h];
    for (int l = 0; l < LAT; ++l) t = fmaf(z[b * LAT + l], Wd1[h * LAT + l], t);
    hg1[i] = fmaxf(t, 0.0f);
}
__global__ void k_hg2(const float* __restrict__ hg1, const float* __restrict__ Wd2,
                      const float* __restrict__ bd2, float* __restrict__ hg) {
    int i = blockIdx.x * blockDim.x + threadIdx.x;
    if (i >= Bsz * HID) return;
    int b = i >> 7, h = i & 127;
    float t = bd2[h];
    for (int k = 0; k < HID; ++k) t = fmaf(hg1[b * HID + k], Wd2[h * HID + k], t);
    hg[i] = t;
}

// ---------------- hn = x@Wnp^T + bnp + hg[b] ----------------
__global__ void k_hnfin(const float* __restrict__ hnp, const float* __restrict__ bnp,
                        const float* __restrict__ hg, float* __restrict__ hn) {
    int i = blockIdx.x * blockDim.x + threadIdx.x;
    if (i >= NT * HID) return;
    int n = i >> 7, c = i & 127;
    hn[i] = hnp[i] + bnp[c] + hg[(n >> 8) * HID + c];
}

// ---------------- adj ground truth + mask ----------------
__global__ void k_adjmask_init(float* __restrict__ out) {
    int i = blockIdx.x * blockDim.x + threadIdx.x;
    if (i < BNN) out[BNN + i] = 0.0f;                         // adj_gt zeroed
    else if (i < BNN + Bsz * Nn) out[2 * BNN + (i - BNN)] = 1.0f; // mask = True
}
__global__ void k_adj(const int* __restrict__ src, const int* __restrict__ dst,
                      float* __restrict__ out) {
    int e = blockIdx.x * blockDim.x + threadIdx.x;
    if (e >= NE) return;
    int s = src[e], d = dst[e];
    int gid = s >> 8;                                          // s / Nn
    atomicAdd(&out[BNN + gid * (Nn * Nn) + (s & 255) * Nn + (d & 255)], 1.0f);
}

// ---------------- fused edge decoder: logits[b,i,j] ----------------
// Never materializes the [B,N,N,HID] tensor. 16x16 (i,j) tile per block,
// pa/pb tiles staged in LDS with stride 129 (bank-conflict free).
__global__ void k_edge_dec(const float* __restrict__ pa, const float* __restrict__ pb,
                           const float* __restrict__ be1, const float* __restrict__ We2,
                           const float* __restrict__ be2, float* __restrict__ logits) {
    __shared__ float paS[16 * 129];
    __shared__ float pbS[16 * 129];
    __shared__ float wS[HID];
    __shared__ float beS[HID];
    const int b = blockIdx.z, it = blockIdx.y, jt = blockIdx.x;
    const int t = threadIdx.x;                 // 256 threads
#pragma unroll
    for (int r = 0; r < 8; ++r) {              // 2048 elements per tile
        int e = r * 256 + t;
        int row = e >> 7, k = e & 127;
        paS[row * 129 + k] = pa[(b * Nn + it * 16 + row) * HID + k];
        pbS[row * 129 + k] = pb[(b * Nn + jt * 16 + row) * HID + k];
    }
    if (t < HID) { wS[t] = We2[t]; beS[t] = be1[t]; }
    __syncthreads();
    const int ti = t & 15, tj = t >> 4;
    float acc = be2[0];
#pragma unroll 4
    for (int k = 0; k < HID; ++k) {
        float v = paS[ti * 129 + k] + pbS[tj * 129 + k] + beS[k];
        acc = fmaf(fmaxf(v, 0.0f), wS[k], acc);
    }
    logits[b * (Nn * Nn) + (it * 16 + ti) * Nn + (jt * 16 + tj)] = acc;
}

// =====================================================================
extern "C" void kernel_launch(void* const* d_in, const int* in_sizes, int n_in,
                              void* d_out, int out_size, void* d_ws, size_t ws_size,
                              hipStream_t stream) {
    (void)in_sizes; (void)n_in; (void)out_size; (void)ws_size;
    const float* x   = (const float*)d_in[0];
    const float* eps = (const float*)d_in[1];
    const float* W1  = (const float*)d_in[2];  const float* b1  = (const float*)d_in[3];
    const float* g1  = (const float*)d_in[4];  const float* bb1 = (const float*)d_in[5];
    const float* W2  = (const float*)d_in[6];  const float* b2  = (const float*)d_in[7];
    const float* g2  = (const float*)d_in[8];  const float* bb2 = (const float*)d_in[9];
    const float* W3  = (const float*)d_in[10]; const float* b3  = (const float*)d_in[11];
    const float* g3  = (const float*)d_in[12]; const float* bb3 = (const float*)d_in[13];
    const float* Wnp = (const float*)d_in[14]; const float* bnp = (const float*)d_in[15];
    const float* Wd1 = (const float*)d_in[16]; const float* bd1 = (const float*)d_in[17];
    const float* Wd2 = (const float*)d_in[18]; const float* bd2 = (const float*)d_in[19];
    const float* We1 = (const float*)d_in[20]; const float* be1 = (const float*)d_in[21];
    const float* We2 = (const float*)d_in[22]; const float* be2 = (const float*)d_in[23];
    const float* Wmu = (const float*)d_in[24]; const float* bmu = (const float*)d_in[25];
    const float* Wlv = (const float*)d_in[26]; const float* blv = (const float*)d_in[27];
    const int* esrc  = (const int*)d_in[28];
    const int* edst  = (const int*)d_in[29];

    float* out = (float*)d_out;
    // output layout: logits[0..BNN) | adj[BNN..2BNN) | mask | mu | logvar
    float* mu_out = out + 2 * BNN + Bsz * Nn;
    float* lv_out = mu_out + Bsz * LAT;

    // workspace layout (floats)
    float* ws   = (float*)d_ws;
    float* dinv = ws;                    // 4096
    float* H    = dinv + NT;             // 524288  (gemm out; later pa)
    float* AGG  = H + NT * HID;          // 524288  (aggregation; later pb)
    float* BUFA = AGG + NT * HID;        // 524288  (bn out; later hn)
    float* gbuf = BUFA + NT * HID;       // 2048
    float* zbuf = gbuf + Bsz * HID;      // 512
    float* hg1b = zbuf + Bsz * LAT;      // 2048
    float* hgb  = hg1b + Bsz * HID;      // 2048

    const dim3 gemm_grid(HID / 16, NT / 16);   // (8, 256) tiles

    // --- degree normalization ---
    k_deg_init<<<NT / 256, 256, 0, stream>>>(dinv);
    k_deg_acc<<<NE / 256, 256, 0, stream>>>(edst, dinv);
    k_dinv<<<NT / 256, 256, 0, stream>>>(dinv);

    // --- GCN layer 1 (K=64) ---
    gemm_wmma_f32<<<gemm_grid, 32, 0, stream>>>(x, INF_, W1, INF_, 0, H, HID, INF_);
    k_agg_init<<<NT * HID / 256, 256, 0, stream>>>(H, dinv, b1, AGG);
    k_agg_edges<<<NE, HID, 0, stream>>>(H, dinv, esrc, edst, AGG);
    k_bn_lrelu<<<HID, 256, 0, stream>>>(AGG, g1, bb1, BUFA);
    // --- GCN layer 2 (K=128) ---
    gemm_wmma_f32<<<gemm_grid, 32, 0, stream>>>(BUFA, HID, W2, HID, 0, H, HID, HID);
    k_agg_init<<<NT * HID / 256, 256, 0, stream>>>(H, dinv, b2, AGG);
    k_agg_edges<<<NE, HID, 0, stream>>>(H, dinv, esrc, edst, AGG);
    k_bn_lrelu<<<HID, 256, 0, stream>>>(AGG, g2, bb2, BUFA);
    // --- GCN layer 3 ---
    gemm_wmma_f32<<<gemm_grid, 32, 0, stream>>>(BUFA, HID, W3, HID, 0, H, HID, HID);
    k_agg_init<<<NT * HID / 256, 256, 0, stream>>>(H, dinv, b3, AGG);
    k_agg_edges<<<NE, HID, 0, stream>>>(H, dinv, esrc, edst, AGG);
    k_bn_lrelu<<<HID, 256, 0, stream>>>(AGG, g3, bb3, BUFA);

    // --- latent head ---
    k_pool<<<Bsz * HID / 256, 256, 0, stream>>>(BUFA, gbuf);
    k_muz<<<(Bsz * LAT + 255) / 256, 256, 0, stream>>>(gbuf, Wmu, bmu, Wlv, blv, eps,
                                                       mu_out, lv_out, zbuf);
    k_hg1<<<Bsz * HID / 256, 256, 0, stream>>>(zbuf, Wd1, bd1, hg1b);
    k_hg2<<<Bsz * HID / 256, 256, 0, stream>>>(hg1b, Wd2, bd2, hgb);

    // --- node projection: hn = x@Wnp^T + bnp + hg[b] ---
    gemm_wmma_f32<<<gemm_grid, 32, 0, stream>>>(x, INF_, Wnp, INF_, 0, H, HID, INF_);
    k_hnfin<<<NT * HID / 256, 256, 0, stream>>>(H, bnp, hgb, BUFA);

    // --- pa / pb from split We1 (row stride 2*HID) ---
    gemm_wmma_f32<<<gemm_grid, 32, 0, stream>>>(BUFA, HID, We1, 2 * HID, 0,   H,   HID, HID);
    gemm_wmma_f32<<<gemm_grid, 32, 0, stream>>>(BUFA, HID, We1, 2 * HID, HID, AGG, HID, HID);

    // --- adjacency ground truth + mask ---
    k_adjmask_init<<<(BNN + Bsz * Nn + 255) / 256, 256, 0, stream>>>(out);
    k_adj<<<NE / 256, 256, 0, stream>>>(esrc, edst, out);

    // --- fused edge decoder (dominant kernel) ---
    k_edge_dec<<<dim3(Nn / 16, Nn / 16, Bsz), 256, 0, stream>>>(H, AGG, be1, We2, be2, out);
}